// AlgoMiniBatch_57844619542864
// MI455X (gfx1250) — compile-verified
//
#include <hip/hip_runtime.h>
#include <hip/hip_bf16.h>
#include <math.h>

typedef float v2f __attribute__((ext_vector_type(2)));
typedef float v8f __attribute__((ext_vector_type(8)));

#define D128 128
#define K256 256
#define B_SZ 1024
#define S1_F 25
#define S2_F 10
#define M1 (B_SZ + B_SZ * S2_F)     // 11264 rows for layer-1 GEMM
#define LDS_STRIDE 264              // 256 + pad (keeps rows on distinct banks)

// ---------------------------------------------------------------------------
// Kernel 1: build A1[r][0:128] = x[self(r)], A1[r][128:256] = mean_s x[nb(r,s)]
// One 64-thread block per row; float4 (16B) lanes -> 512B coalesced row loads.
// ---------------------------------------------------------------------------
__global__ __launch_bounds__(64)
void gather_concat1(const float* __restrict__ x,
                    const int* __restrict__ nodes,
                    const int* __restrict__ nb2,
                    const int* __restrict__ nb1_self,
                    const int* __restrict__ nb1_nb,
                    float* __restrict__ A1)
{
    const int r = blockIdx.x;          // 0 .. M1-1
    const int t = threadIdx.x;         // 0 .. 63
    int selfIdx;
    const int* aggIdx;
    if (r < B_SZ) {
        selfIdx = nodes[r];
        aggIdx  = nb1_self + r * S1_F;
    } else {
        const int q = r - B_SZ;        // flat (b*S2 + j)
        selfIdx = nb2[q];
        aggIdx  = nb1_nb + q * S1_F;
    }
    const float4* xv  = (const float4*)x;                   // row stride = 32
    float4*       out = (float4*)(A1 + (size_t)r * K256);   // 64 float4 / row
    if (t < 32) {
        out[t] = xv[(size_t)selfIdx * 32 + t];
    } else {
        const int c = t - 32;
        float4 acc = make_float4(0.f, 0.f, 0.f, 0.f);
        #pragma unroll
        for (int s = 0; s < S1_F; ++s) {
            const int idx = aggIdx[s];
            const float4 v = xv[(size_t)idx * 32 + c];
            acc.x += v.x; acc.y += v.y; acc.z += v.z; acc.w += v.w;
        }
        const float inv = 1.0f / (float)S1_F;
        acc.x *= inv; acc.y *= inv; acc.z *= inv; acc.w *= inv;
        out[32 + c] = acc;
    }
}

// ---------------------------------------------------------------------------
// Kernel 2: build A2[b][0:128] = H1[b], A2[b][128:256] = mean_j H1[1024+b*10+j]
// ---------------------------------------------------------------------------
__global__ __launch_bounds__(64)
void gather_concat2(const float* __restrict__ H1, float* __restrict__ A2)
{
    const int b = blockIdx.x;
    const int t = threadIdx.x;
    const float4* hv  = (const float4*)H1;                  // row stride = 32
    float4*       out = (float4*)(A2 + (size_t)b * K256);
    if (t < 32) {
        out[t] = hv[(size_t)b * 32 + t];
    } else {
        const int c = t - 32;
        float4 acc = make_float4(0.f, 0.f, 0.f, 0.f);
        #pragma unroll
        for (int j = 0; j < S2_F; ++j) {
            const float4 v = hv[((size_t)(B_SZ + b * S2_F + j)) * 32 + c];
            acc.x += v.x; acc.y += v.y; acc.z += v.z; acc.w += v.w;
        }
        const float inv = 1.0f / (float)S2_F;
        acc.x *= inv; acc.y *= inv; acc.z *= inv; acc.w *= inv;
        out[32 + c] = acc;
    }
}

// ---------------------------------------------------------------------------
// Kernel 3: fused  out = l2norm(relu(A @ W + bias))   via V_WMMA_F32_16X16X4_F32
//   A:[M,256]  W:[256,128]  bias:[128]  out:[M,128]
//   block = 256 threads = 8 wave32; block -> 16-row M slab (LDS-staged),
//   wave w -> 16-column N tile; 64 WMMA steps over K=256.
// ---------------------------------------------------------------------------
__global__ __launch_bounds__(256)
void sage_gemm_norm(const float* __restrict__ A,
                    const float* __restrict__ W,
                    const float* __restrict__ bias,
                    float* __restrict__ out)
{
    __shared__ float As[16 * LDS_STRIDE];   // 16 x 256 A-tile, padded
    __shared__ float red[16][8];            // per-row, per-wave sum-of-squares

    const int m0  = blockIdx.x * 16;
    const int tid = threadIdx.x;

    // Cooperative A-tile load: 4096 floats, 16 per thread, coalesced columns.
    #pragma unroll
    for (int i = 0; i < 16; ++i) {
        const int flat = tid + i * 256;
        const int row  = flat >> 8;
        const int col  = flat & 255;
        As[row * LDS_STRIDE + col] = A[(size_t)(m0 + row) * K256 + col];
    }
    __syncthreads();

    const int lane  = tid & 31;
    const int wave  = tid >> 5;      // 0..7 -> N tile
    const int l16   = lane & 15;
    const int hi    = lane >> 4;     // 0 or 1
    const int khalf = hi * 2;        // K sub-pair within the 16x16x4 step
    const int nb    = wave * 16;

    v8f c = {};
    const float* wcol = W + nb + l16;           // column n = nb + l16
    const float* arow = As + l16 * LDS_STRIDE;  // row m = l16

    for (int k = 0; k < K256; k += 4) {
        v2f a, b;
        const float* ap = arow + k + khalf;
        a.x = ap[0];
        a.y = ap[1];
        const float* bp = wcol + (size_t)(k + khalf) * D128;
        b.x = bp[0];
        b.y = bp[D128];
        c = __builtin_amdgcn_wmma_f32_16x16x4_f32(false, a, false, b,
                                                  (short)0, c, false, false);
    }

    // Epilogue: bias + ReLU, then row-wise L2 normalization.
    const float bv = bias[nb + l16];
    float cv[8], ss[8];
    #pragma unroll
    for (int i = 0; i < 8; ++i) {
        cv[i] = fmaxf(c[i] + bv, 0.0f);
        ss[i] = cv[i] * cv[i];
    }
    // Reduce over the 16 columns held by this wave (within 16-lane groups).
    #pragma unroll
    for (int off = 1; off < 16; off <<= 1) {
        #pragma unroll
        for (int i = 0; i < 8; ++i)
            ss[i] += __shfl_xor(ss[i], off, 32);
    }
    if (l16 == 0) {
        #pragma unroll
        for (int i = 0; i < 8; ++i)
            red[hi * 8 + i][wave] = ss[i];
    }
    __syncthreads();

    #pragma unroll
    for (int i = 0; i < 8; ++i) {
        const int r = hi * 8 + i;               // row within the 16-row slab
        float s = 0.0f;
        #pragma unroll
        for (int w = 0; w < 8; ++w) s += red[r][w];
        const float inv = (s > 0.0f) ? rsqrtf(s) : 1.0f;
        out[(size_t)(m0 + r) * D128 + nb + l16] = cv[i] * inv;
    }
}

// ---------------------------------------------------------------------------
// Host-side launch: gather1 -> gemm1 -> gather2 -> gemm2, all on `stream`.
// ---------------------------------------------------------------------------
extern "C" void kernel_launch(void* const* d_in, const int* in_sizes, int n_in,
                              void* d_out, int out_size, void* d_ws, size_t ws_size,
                              hipStream_t stream)
{
    (void)in_sizes; (void)n_in; (void)out_size; (void)ws_size;

    const float* x       = (const float*)d_in[0];   // [500000,128]
    const float* W1      = (const float*)d_in[1];   // [256,128]
    const float* b1      = (const float*)d_in[2];   // [128]
    const float* W2      = (const float*)d_in[3];   // [256,128]
    const float* b2      = (const float*)d_in[4];   // [128]
    const int*   nodes   = (const int*)d_in[5];     // [1024]
    const int*   nb2     = (const int*)d_in[6];     // [1024,10]
    const int*   nb1s    = (const int*)d_in[7];     // [1024,25]
    const int*   nb1n    = (const int*)d_in[8];     // [1024,10,25]
    float*       zout    = (float*)d_out;           // [1024,128]

    // Workspace layout (fp32), ~18.3 MB total:
    float* A1 = (float*)d_ws;                       // [11264,256]
    float* H1 = A1 + (size_t)M1 * K256;             // [11264,128]
    float* A2 = H1 + (size_t)M1 * D128;             // [1024,256]

    // 1) Layer-1 gather + mean-concat for targets and layer-2 neighbors.
    gather_concat1<<<M1, 64, 0, stream>>>(x, nodes, nb2, nb1s, nb1n, A1);

    // 2) Layer-1 fused GEMM + ReLU + L2 norm  -> H1 (targets rows 0..1023,
    //    neighbor embeddings rows 1024..11263).
    sage_gemm_norm<<<M1 / 16, 256, 0, stream>>>(A1, W1, b1, H1);

    // 3) Layer-2 concat: self embedding + mean over 10 neighbor embeddings.
    gather_concat2<<<B_SZ, 64, 0, stream>>>(H1, A2);

    // 4) Layer-2 fused GEMM + ReLU + L2 norm -> final output.
    sage_gemm_norm<<<B_SZ / 16, 256, 0, stream>>>(A2, W2, b2, zout);
}